// GraphNeighborhoodSampler_23570780521124
// MI455X (gfx1250) — compile-verified
//
#include <hip/hip_runtime.h>
#include <math.h>

// Problem constants (from the reference)
#define NPTS   6000
#define KTOT   1800          // K
#define KNN    1620          // K - NUM_RND
#define NRND   180           // NUM_RND
#define NPAD   8192          // next pow2 >= NPTS for bitonic sort
#define TILES  375           // NPTS / 16
#define NGRP   47            // ceil(TILES / 8): column-tile groups per wave
#define MASKW  188           // ceil(6000 / 32)
#define CANDM  4379          // c - 1 = (N - KNN) - 1 : modulus of the position permutation

typedef float v2f __attribute__((ext_vector_type(2)));
typedef float v8f __attribute__((ext_vector_type(8)));
typedef int   v4i_b128 __attribute__((vector_size(4 * sizeof(int))));  // matches builtin param

#if defined(__has_builtin)
#if __has_builtin(__builtin_amdgcn_global_load_async_to_lds_b128)
#define HAVE_ASYNC_LDS 1
#endif
#endif
#ifndef HAVE_ASYNC_LDS
#define HAVE_ASYNC_LDS 0
#endif

// ---------------------------------------------------------------------------
// Kernel 1: pairwise squared distances via V_WMMA_F32_16X16X4_F32.
// Each wave owns one 16-row tile strip and sweeps 8 column tiles: A (row
// coords) and the per-row sq values are loaded once and reused across 8 WMMAs.
// A (16x4 f32): lanes 0-15 hold M=lane, v[0]=K0(x), v[1]=K1(y); lanes 16-31
// hold K=2,3 which are zero-padded (D=2). B (4x16) mirrored. C/D 16x16 f32:
// VGPR r -> M = r + 8*(lane>=16), N = lane&15.
// ---------------------------------------------------------------------------
__global__ __launch_bounds__(256)
void gns_dist_wmma(const float* __restrict__ coords, float* __restrict__ d2out) {
  const int wave = (int)((blockIdx.x * blockDim.x + threadIdx.x) >> 5);
  if (wave >= TILES * NGRP) return;               // wave-uniform: EXEC stays all-1
  const int tr   = wave / NGRP;
  const int grp  = wave % NGRP;
  const int lane = (int)(threadIdx.x & 31);
  const int half = lane >> 4;
  const int l16  = lane & 15;
  const float2* c2 = (const float2*)coords;

  // A matrix + per-row sq, invariant across the 8 column tiles.
  float2 ar = c2[tr * 16 + l16];
  v2f a;
  a[0] = half ? 0.0f : ar.x;                      // K=0 (x); lanes 16-31 are K=2,3 -> 0
  a[1] = half ? 0.0f : ar.y;                      // K=1 (y)
  float sqm[8];
#pragma unroll
  for (int r = 0; r < 8; ++r) {
    float2 cm = c2[tr * 16 + r + half * 8];
    sqm[r] = cm.x * cm.x + cm.y * cm.y;
  }

#pragma unroll
  for (int u = 0; u < 8; ++u) {
    const int tc = grp * 8 + u;
    if (tc >= TILES) break;                       // wave-uniform

    float2 bc = c2[tc * 16 + l16];
    v2f b;
    b[0] = half ? 0.0f : bc.x;
    b[1] = half ? 0.0f : bc.y;

    v8f acc = {};
    acc = __builtin_amdgcn_wmma_f32_16x16x4_f32(
        /*neg_a=*/false, a, /*neg_b=*/false, b,
        /*c_mod=*/(short)0, acc, /*reuse_a=*/false, /*reuse_b=*/false);

    const int ncol = tc * 16 + l16;
    float2 cn = c2[ncol];
    const float sqn = cn.x * cn.x + cn.y * cn.y;
#pragma unroll
    for (int r = 0; r < 8; ++r) {
      const int nrow = tr * 16 + r + half * 8;
      const float d2 = fmaxf(sqm[r] + sqn - 2.0f * acc[r], 0.0f);  // clamp keeps key monotonic
      d2out[(size_t)nrow * NPTS + ncol] = d2;
    }
  }
}

// ---------------------------------------------------------------------------
// Kernel 2: per-row top-KNN (sorted, index tie-break) via LDS bitonic sort of
// packed u64 keys, then negative sampling via bitmask rank-select. The mask
// and prefix tables live in the +inf-pad region of the sorted array. When the
// d2 matrix is in the workspace, the row is staged into LDS with
// GLOBAL_LOAD_ASYNC_TO_LDS_B128 (ASYNCcnt path), snapshotted to registers,
// then the same LDS is overwritten with the packed sort keys.
// ---------------------------------------------------------------------------
__global__ __launch_bounds__(256)
void gns_topk(const float* __restrict__ coords,
              const float* __restrict__ d2mat, int use_ws,
              float* __restrict__ out) {
  __shared__ unsigned long long buf[NPAD];        // 64 KB LDS
  const int row = (int)blockIdx.x;
  const int tid = (int)threadIdx.x;
  const float2* c2 = (const float2*)coords;
  const float2 ci = c2[row];

  if (use_ws) {
#if HAVE_ASYNC_LDS
    // Stage the 24000-byte d2 row into the low 24 KB of buf via async copy.
    float* stage = (float*)buf;
    const float* rowp = d2mat + (size_t)row * NPTS;
    for (int e = tid; e < NPTS / 4; e += 256) {   // 1500 x b128
      __builtin_amdgcn_global_load_async_to_lds_b128(
          (v4i_b128*)(rowp + e * 4), (v4i_b128*)(stage + e * 4), 0, 0);
    }
#if __has_builtin(__builtin_amdgcn_s_wait_asynccnt)
    __builtin_amdgcn_s_wait_asynccnt(0);
#else
    asm volatile("s_wait_asynccnt 0x0" ::: "memory");
#endif
    __syncthreads();
    // Snapshot to registers (24 per thread covers 6144 >= 6000), then overwrite
    // the staging region with packed (f32 bits << 32) | index sort keys.
    float vals[24];
#pragma unroll
    for (int s = 0; s < 24; ++s) {
      const int j = tid + s * 256;
      vals[s] = (j < NPTS) ? stage[j] : 0.0f;
    }
    __syncthreads();
#pragma unroll
    for (int s = 0; s < 24; ++s) {
      const int j = tid + s * 256;
      if (j < NPTS) {
        const unsigned key = __float_as_uint(fmaxf(vals[s], 0.0f));
        buf[j] = ((unsigned long long)key << 32) | (unsigned)j;
      }
    }
    for (int j = NPTS + tid; j < NPAD; j += 256)
      buf[j] = (0xFFFFFFFFull << 32) | (unsigned)j;
#else
    for (int j = tid; j < NPAD; j += 256) {
      unsigned key = 0xFFFFFFFFu;
      if (j < NPTS) key = __float_as_uint(fmaxf(d2mat[(size_t)row * NPTS + j], 0.0f));
      buf[j] = ((unsigned long long)key << 32) | (unsigned)j;
    }
#endif
  } else {
    // No workspace: recompute d2 directly (coords are L2-resident, ~free).
    for (int j = tid; j < NPAD; j += 256) {
      unsigned key = 0xFFFFFFFFu;
      if (j < NPTS) {
        float2 cj = c2[j];
        const float dx = ci.x - cj.x, dy = ci.y - cj.y;
        key = __float_as_uint(fmaxf(dx * dx + dy * dy, 0.0f));
      }
      buf[j] = ((unsigned long long)key << 32) | (unsigned)j;
    }
  }
  __syncthreads();

  // Bitonic sort ascending (91 stages for 8192).
  for (int k = 2; k <= NPAD; k <<= 1) {
    for (int jj = k >> 1; jj > 0; jj >>= 1) {
      for (int i = tid; i < NPAD; i += 256) {
        const int l = i ^ jj;
        if (l > i) {
          unsigned long long x = buf[i], y = buf[l];
          const bool up = ((i & k) == 0);
          if ((x > y) == up) { buf[i] = y; buf[l] = x; }
        }
      }
      __syncthreads();
    }
  }

  const size_t NK = (size_t)NPTS * KTOT;
  float* e0  = out;                 // edge_idx[0] : neighbor / negative
  float* e1  = out + NK;            // edge_idx[1] : target (row)
  float* wgt = out + 2 * NK;        // edge weights

  // Emit kNN edges (sorted ascending distance; entry 0 is self, d=0).
  for (int j = tid; j < KNN; j += 256) {
    const unsigned idx = (unsigned)buf[j];
    const size_t o = (size_t)row * KTOT + j;
    e0[o] = (float)idx;
    e1[o] = (float)row;
    float2 cj = c2[idx];
    const float dx = cj.x - ci.x, dy = cj.y - ci.y;
    wgt[o] = sqrtf(dx * dx + dy * dy);
  }

  // Non-neighbor bitmask + popcount prefix, in the pad region (buf[6000..]).
  unsigned* mask = (unsigned*)&buf[7168];         // 188 u32
  unsigned* pref = (unsigned*)&buf[7424];         // 189 u32
  __syncthreads();
  for (int w = tid; w < MASKW; w += 256)
    mask[w] = (w == MASKW - 1) ? 0x0000FFFFu : 0xFFFFFFFFu;  // clear bits >= 6000
  __syncthreads();
  for (int j = tid; j < KNN; j += 256) {
    const unsigned idx = (unsigned)buf[j];
    atomicAnd(&mask[idx >> 5], ~(1u << (idx & 31)));
  }
  __syncthreads();
  if (tid == 0) {
    unsigned s = 0;
    for (int w = 0; w < MASKW; ++w) { pref[w] = s; s += (unsigned)__popc(mask[w]); }
    pref[MASKW] = s;                // total = 4380 non-neighbors
  }
  __syncthreads();

  // 180 negatives: fixed affine position permutation mod (c-1)=4379, then
  // rank-select the p-th set bit (ascending candidate order, as in reference).
  if (tid < NRND) {
    int p = (int)(((long)tid * 24407 + 1013) % CANDM);   // in [0, 4378] < 4380
    int lo = 0, hi = MASKW - 1;
    while (lo < hi) {
      const int mid = (lo + hi + 1) >> 1;
      if ((int)pref[mid] <= p) lo = mid; else hi = mid - 1;
    }
    int r = p - (int)pref[lo];
    unsigned m = mask[lo];
    int bit = 0;
    for (;;) {
      bit = __ffs(m) - 1;
      if (r == 0) break;
      m &= m - 1;
      --r;
    }
    const int idx = lo * 32 + bit;
    const size_t o = (size_t)row * KTOT + KNN + tid;
    e0[o] = (float)idx;
    e1[o] = (float)row;
    float2 cj = c2[idx];
    const float dx = cj.x - ci.x, dy = cj.y - ci.y;
    wgt[o] = sqrtf(dx * dx + dy * dy);
  }

  if (row == 0 && tid == 0) out[3 * NK] = (float)KTOT;   // scalar K
}

// ---------------------------------------------------------------------------
extern "C" void kernel_launch(void* const* d_in, const int* in_sizes, int n_in,
                              void* d_out, int out_size, void* d_ws, size_t ws_size,
                              hipStream_t stream) {
  (void)in_sizes; (void)n_in; (void)out_size;
  const float* coords = (const float*)d_in[0];
  float* out = (float*)d_out;

  const size_t need = (size_t)NPTS * NPTS * sizeof(float);   // 144 MB d2 matrix
  const int use_ws = (ws_size >= need) ? 1 : 0;

  if (use_ws) {
    const int totalWaves = TILES * NGRP;                     // 17625 waves, 8 tiles each
    const int blocks = (totalWaves + 7) / 8;                 // 8 waves / 256-thread block
    gns_dist_wmma<<<blocks, 256, 0, stream>>>(coords, (float*)d_ws);
  }
  gns_topk<<<NPTS, 256, 0, stream>>>(coords, (const float*)d_ws, use_ws, out);
}